// GCNNet_NC_12257836663288
// MI455X (gfx1250) — compile-verified
//
#include <hip/hip_runtime.h>
#include <hip/hip_bf16.h>

// ---------------------------------------------------------------------------
// GCN (3x GCNConv + ReLU) + prototype-distance head on gfx1250 (MI455X).
//  - Dense GEMMs: V_WMMA_F32_16X16X4_F32 (full fp32 precision, CDNA5 WMMA)
//  - Weight matrix staged into LDS via the Tensor Data Mover
//    (tensor_load_to_lds + s_wait_tensorcnt), D# built per ISA sec. 8
//  - Sparse aggregation: native global_atomic_add_f32 on L2-resident rows
// ---------------------------------------------------------------------------

typedef float v2f __attribute__((ext_vector_type(2)));
typedef float v8f __attribute__((ext_vector_type(8)));
typedef unsigned int v4u __attribute__((ext_vector_type(4)));
typedef int v4i __attribute__((ext_vector_type(4)));
typedef int v8i __attribute__((ext_vector_type(8)));

#define FDIM 128   // hidden width of every layer (reference fixes this)
#define PPAD 64    // prototypes (50) padded to a multiple of 16
#define NPROTO 50
#define NCLASS 10

// ---------------- degree / normalization --------------------------------

__global__ void fill_ones(float* __restrict__ p, int n) {
    int i = blockIdx.x * blockDim.x + threadIdx.x;
    if (i < n) p[i] = 1.0f;   // self-loop contributes 1 to every node's degree
}

__global__ void deg_accum(const int* __restrict__ ei, float* __restrict__ deg,
                          int E) {
    int e = blockIdx.x * blockDim.x + threadIdx.x;
    if (e < E) {
        int dst = ei[E + e];                 // edge_index[1][e]
        unsafeAtomicAdd(&deg[dst], 1.0f);    // counts are small ints: exact
    }
}

__global__ void deg_to_rsqrt(float* __restrict__ dis, int n) {
    int i = blockIdx.x * blockDim.x + threadIdx.x;
    if (i < n) dis[i] = __frsqrt_rn(dis[i]); // deg >= 1 always (self-loop)
}

// ---------------- fp32 WMMA GEMM:  C[M,Fout] = A[M,K] * W[Fout,K]^T ------
// One wave computes a 16-row block x (NT*16) output columns.
// A-fragment (16x4 f32, 2 VGPR):  lane L holds A[M=L&15, K = k0+2*(L>>4)+{0,1}]
// B-fragment ( 4x16 f32, 2 VGPR): lane L holds B[K = k0+2*(L>>4)+{0,1}, N=L&15]
//   with B = W^T  =>  contiguous float2 from row-major W, staged in LDS by TDM.
// C/D (16x16 f32, 8 VGPR): VGPR i, lane L -> C[M = i+8*(L>>4), N = L&15].

template <int NT, int KCH>
__global__ __launch_bounds__(256) void wmma_gemm_xWt(
    const float* __restrict__ A, const float* __restrict__ W,
    float* __restrict__ C, int M, int K) {
    __shared__ float Wlds[NT * 16 * KCH];   // [Fout][KCH] fp32 chunk of W

    const int wave = threadIdx.x >> 5;
    const int lane = threadIdx.x & 31;
    const int row0 = (blockIdx.x * 8 + wave) * 16;
    // wave 0 always owns a valid tile (it has the smallest row0 in the block);
    // other waves may terminate early — terminated waves don't block barriers.
    if (row0 >= M) return;

    const int half = lane >> 4;               // 0 or 1
    const int l16  = lane & 15;
    const int Fout = NT * 16;

    v8f acc[NT];
    const v8f vzero = {0.f, 0.f, 0.f, 0.f, 0.f, 0.f, 0.f, 0.f};
#pragma unroll
    for (int t = 0; t < NT; ++t) acc[t] = vzero;

    const float* aBase = A + (size_t)(row0 + l16) * K + 2 * half;
    const unsigned lds_off = (unsigned)(size_t)(&Wlds[0]); // low 32b = LDS offset

    for (int k0 = 0; k0 < K; k0 += KCH) {
        __syncthreads();                      // previous chunk fully consumed
        if (wave == 0) {
            // ---- Tensor DMA descriptor (ISA sec. 8): 2-D tile KCH x Fout ----
            unsigned long long ga =
                (unsigned long long)(size_t)(W + k0);   // tile start (bytes)
            v4u g0;
            g0.x = 1u;                                   // count=1, user D#
            g0.y = lds_off;                              // LDS dest (bytes)
            g0.z = (unsigned)(ga & 0xFFFFFFFFull);       // global_addr[31:0]
            g0.w = (unsigned)((ga >> 32) & 0x1FFFFFFull) // global_addr[56:32]
                   | (2u << 30);                         // type = 2 ("image")
            int rem = K - k0;                            // cols left in tensor
            v8i g1;
            g1[0] = 0x00020000;                          // data_size = 4 bytes
            g1[1] = (rem & 0xFFFF) << 16;                // tensor_dim0[15:0]
            g1[2] = ((unsigned)rem >> 16) | ((Fout & 0xFFFF) << 16); // dim0 hi | dim1 lo
            g1[3] = (KCH << 16);                         // tile_dim0 = KCH
            g1[4] = Fout;                                // tile_dim1 = Fout rows
            g1[5] = K;                                   // tensor_dim0_stride
            g1[6] = 0;
            g1[7] = 0;
            v4i gz4 = {0, 0, 0, 0};                      // <=2-D: groups 2/3 unused
            v8i gz8 = {0, 0, 0, 0, 0, 0, 0, 0};          // unused trailing group
            __builtin_amdgcn_tensor_load_to_lds(g0, g1, gz4, gz4, gz8, 0);
            __builtin_amdgcn_s_wait_tensorcnt(0);
        }
        __syncthreads();                      // chunk visible to all waves

#pragma unroll 4
        for (int kk = 0; kk < KCH; kk += 4) {
            const float2 at = *(const float2*)(aBase + k0 + kk);
            v2f a; a.x = at.x; a.y = at.y;
#pragma unroll
            for (int t = 0; t < NT; ++t) {
                const float2 bt = *(const float2*)(
                    &Wlds[(t * 16 + l16) * KCH + kk + 2 * half]);
                v2f b; b.x = bt.x; b.y = bt.y;
                acc[t] = __builtin_amdgcn_wmma_f32_16x16x4_f32(
                    false, a, false, b, (short)0, acc[t], false, false);
            }
        }
    }

#pragma unroll
    for (int t = 0; t < NT; ++t) {
#pragma unroll
        for (int i = 0; i < 8; ++i) {
            C[(size_t)(row0 + i + 8 * half) * Fout + t * 16 + l16] = acc[t][i];
        }
    }
}

// ---------------- sparse aggregation -------------------------------------

// agg[n] = dis[n]^2 * m[n]   (self-loop folded into the init pass)
__global__ void agg_init(const float* __restrict__ m,
                         const float* __restrict__ dis,
                         float* __restrict__ agg, int N) {
    size_t idx = (size_t)blockIdx.x * blockDim.x + threadIdx.x;
    if (idx >= (size_t)N * FDIM) return;
    int n = (int)(idx >> 7);
    float sl = dis[n] * dis[n];
    agg[idx] = sl * m[idx];
}

// agg[dst] += dis[src]*dis[dst] * m[src]   (one wave per edge, 128 features)
__global__ void edge_aggregate(const int* __restrict__ ei,
                               const float* __restrict__ m,
                               const float* __restrict__ dis,
                               float* __restrict__ agg, int E) {
    size_t t = (size_t)blockIdx.x * blockDim.x + threadIdx.x;
    int e = (int)(t >> 5);
    int lane = (int)(t & 31);
    if (e >= E) return;
    const int src = ei[e];
    const int dst = ei[E + e];
    const float nrm = dis[src] * dis[dst];
    const float* mrow = m + (size_t)src * FDIM;
    float* arow = agg + (size_t)dst * FDIM;
#pragma unroll
    for (int j = 0; j < FDIM / 32; ++j) {
        int f = lane + 32 * j;
        unsafeAtomicAdd(&arow[f], nrm * mrow[f]);
    }
}

// h_out = relu(agg + b)
__global__ void layer_finalize(const float* __restrict__ agg,
                               const float* __restrict__ b,
                               float* __restrict__ hout, int N) {
    size_t idx = (size_t)blockIdx.x * blockDim.x + threadIdx.x;
    if (idx >= (size_t)N * FDIM) return;
    int f = (int)(idx & (FDIM - 1));
    hout[idx] = fmaxf(agg[idx] + b[f], 0.0f);
}

// ---------------- prototype head -----------------------------------------

__global__ void proto_prep(const float* __restrict__ protos,
                           float* __restrict__ ppad,
                           float* __restrict__ pnorm) {
    int j = threadIdx.x;  // 64 threads, single block
    if (j >= PPAD) return;
    float s = 0.0f;
    for (int k = 0; k < FDIM; ++k) {
        float v = (j < NPROTO) ? protos[j * FDIM + k] : 0.0f;
        ppad[j * FDIM + k] = v;
        s += v * v;
    }
    pnorm[j] = s;
}

__global__ void proto_post(const float* __restrict__ xp,
                           const float* __restrict__ emb,
                           const float* __restrict__ pnorm,
                           const float* __restrict__ last_w,
                           float* __restrict__ logits,
                           float* __restrict__ probs,
                           float* __restrict__ dist, int N) {
    int n = blockIdx.x * blockDim.x + threadIdx.x;
    if (n >= N) return;

    const float* er = emb + (size_t)n * FDIM;
    float xn = 0.0f;
    for (int k = 0; k < FDIM; ++k) xn += er[k] * er[k];

    float sim[NPROTO];
    for (int j = 0; j < NPROTO; ++j) {
        float d = fmaf(-2.0f, xp[(size_t)n * PPAD + j], xn) + pnorm[j];
        dist[(size_t)n * NPROTO + j] = d;
        sim[j] = __logf((d + 1.0f) / (d + 1e-4f));
    }

    float lg[NCLASS];
    float mx = -1e30f;
    for (int c = 0; c < NCLASS; ++c) {
        float s = 0.0f;
        for (int j = 0; j < NPROTO; ++j) s += sim[j] * last_w[c * NPROTO + j];
        lg[c] = s;
        mx = fmaxf(mx, s);
        logits[(size_t)n * NCLASS + c] = s;
    }
    float se = 0.0f;
    for (int c = 0; c < NCLASS; ++c) {
        lg[c] = __expf(lg[c] - mx);
        se += lg[c];
    }
    float inv = 1.0f / se;
    for (int c = 0; c < NCLASS; ++c)
        probs[(size_t)n * NCLASS + c] = lg[c] * inv;
}

// ---------------- launch ---------------------------------------------------

extern "C" void kernel_launch(void* const* d_in, const int* in_sizes, int n_in,
                              void* d_out, int out_size, void* d_ws,
                              size_t ws_size, hipStream_t stream) {
    const float* x      = (const float*)d_in[0];
    const int*   ei     = (const int*)d_in[1];   // [2,E] int32
    const float* W1     = (const float*)d_in[2];
    const float* b1     = (const float*)d_in[3];
    const float* W2     = (const float*)d_in[4];
    const float* b2     = (const float*)d_in[5];
    const float* W3     = (const float*)d_in[6];
    const float* b3     = (const float*)d_in[7];
    const float* protos = (const float*)d_in[8]; // [50,128]
    const float* last_w = (const float*)d_in[9]; // [10,50]

    const int N  = in_sizes[0] / 512;   // 50000
    const int E  = in_sizes[1] / 2;     // 640000
    const int K1 = in_sizes[2] / FDIM;  // 512

    float* out    = (float*)d_out;
    float* logits = out;                          // [N,10]
    float* probs  = out + (size_t)N * NCLASS;     // [N,10]
    float* emb    = out + (size_t)N * 2 * NCLASS; // [N,128]
    float* dist   = out + (size_t)N * (2 * NCLASS + FDIM); // [N,50]

    float* ws    = (float*)d_ws;
    float* dis   = ws;                            // N
    float* mbuf  = dis + N;                       // N*128
    float* abuf  = mbuf + (size_t)N * FDIM;       // N*128
    float* hbuf  = abuf + (size_t)N * FDIM;       // N*128
    float* xpbuf = hbuf + (size_t)N * FDIM;       // N*64
    float* ppad  = xpbuf + (size_t)N * PPAD;      // 64*128
    float* pnorm = ppad + PPAD * FDIM;            // 64

    const int TB = 256;
    const int gN    = (N + TB - 1) / TB;
    const int gE    = (E + TB - 1) / TB;
    const int gNF   = (int)(((size_t)N * FDIM + TB - 1) / TB);
    const int gEdge = (int)(((size_t)E * 32 + TB - 1) / TB);
    const int gGemm = ((N / 16) + 7) / 8;  // 8 waves (16-row tiles) per block

    // symmetric GCN normalization
    fill_ones<<<gN, TB, 0, stream>>>(dis, N);
    deg_accum<<<gE, TB, 0, stream>>>(ei, dis, E);
    deg_to_rsqrt<<<gN, TB, 0, stream>>>(dis, N);

    // ---- layer 1: 512 -> 128 ----
    wmma_gemm_xWt<8, 64><<<gGemm, TB, 0, stream>>>(x, W1, mbuf, N, K1);
    agg_init<<<gNF, TB, 0, stream>>>(mbuf, dis, abuf, N);
    edge_aggregate<<<gEdge, TB, 0, stream>>>(ei, mbuf, dis, abuf, E);
    layer_finalize<<<gNF, TB, 0, stream>>>(abuf, b1, hbuf, N);

    // ---- layer 2: 128 -> 128 ----
    wmma_gemm_xWt<8, 64><<<gGemm, TB, 0, stream>>>(hbuf, W2, mbuf, N, FDIM);
    agg_init<<<gNF, TB, 0, stream>>>(mbuf, dis, abuf, N);
    edge_aggregate<<<gEdge, TB, 0, stream>>>(ei, mbuf, dis, abuf, E);
    layer_finalize<<<gNF, TB, 0, stream>>>(abuf, b2, hbuf, N);

    // ---- layer 3: 128 -> 128, writes emb straight into d_out ----
    wmma_gemm_xWt<8, 64><<<gGemm, TB, 0, stream>>>(hbuf, W3, mbuf, N, FDIM);
    agg_init<<<gNF, TB, 0, stream>>>(mbuf, dis, abuf, N);
    edge_aggregate<<<gEdge, TB, 0, stream>>>(ei, mbuf, dis, abuf, E);
    layer_finalize<<<gNF, TB, 0, stream>>>(abuf, b3, emb, N);

    // ---- prototype head ----
    proto_prep<<<1, PPAD, 0, stream>>>(protos, ppad, pnorm);
    wmma_gemm_xWt<4, 64><<<gGemm, TB, 0, stream>>>(emb, ppad, xpbuf, N, FDIM);
    proto_post<<<gN, TB, 0, stream>>>(xpbuf, emb, pnorm, last_w, logits, probs,
                                      dist, N);
}